// QuadrotorDynamics_18700287607474
// MI455X (gfx1250) — compile-verified
//
#include <hip/hip_runtime.h>
#include <stdint.h>

// ---------------------------------------------------------------------------
// QuadrotorDynamics on MI455X (gfx1250).
// Memory-bound map: 2M rows x 16 f32 -> 2M rows x 16 f32 (256 MB total,
// ~11 us at 23.3 TB/s).  Both directions of data movement use the CDNA5
// async LDS DMA path:
//   in:  global_load_async_to_lds_b128   (contiguous 16B/lane chunks)
//   out: global_store_async_from_lds_b128 (contiguous 16B/lane chunks)
// with wave-private LDS tiles (counter waits only, zero barriers) and
// 80B-padded LDS rows to minimize bank conflicts.
// ---------------------------------------------------------------------------

typedef float f4 __attribute__((ext_vector_type(4)));

#define THREADS        128          // 4 waves of 32
#define ROWS_PER_BLOCK 128          // one row per thread
#define ROW_F          16           // floats per row
#define PAD_F          20           // padded floats per LDS row (80 B, 16B-aligned)

// Low 32 bits of a generic __shared__ pointer are the LDS byte offset
// (flat aperture scheme).
__device__ __forceinline__ uint32_t lds_off_u32(const void* p) {
    return (uint32_t)(uintptr_t)p;
}

// ----- the dynamics math (mirrors reference f32 expression order) ----------
__device__ __forceinline__ void quad_dynamics(const float xr[16],
                                              const float* __restrict__ a,
                                              const float* __restrict__ b,
                                              const float* __restrict__ c,
                                              float o[16]) {
    const float TL      = (float)(0.707 * 0.17);  // 0.707*L
    const float DARM    = 0.016f;
    const float KT      = 0.1f, KR = 0.1f;
    const float IXX     = 0.007f, IYY = 0.007f, IZZ = 0.012f;
    const float INV_IXX = (float)(1.0 / 0.007);
    const float INV_IYY = (float)(1.0 / 0.007);
    const float INV_IZZ = (float)(1.0 / 0.012);
    const float GZ      = 9.8067f;
    const float MASS    = 0.68f;

    // thrusts = a*u^2 + b*u + c
    const float u0 = xr[12], u1 = xr[13], u2 = xr[14], u3 = xr[15];
    const float t0 = a[0] * u0 * u0 + b[0] * u0 + c[0];
    const float t1 = a[1] * u1 * u1 + b[1] * u1 + c[1];
    const float t2 = a[2] * u2 * u2 + b[2] * u2 + c[2];
    const float t3 = a[3] * u3 * u3 + b[3] * u3 + c[3];

    // torques = TORQUE_MAT @ thrusts
    const float T0 = t0 + t1 + t2 + t3;
    const float T1 = TL * t0 - TL * t1 - TL * t2 + TL * t3;
    const float T2 = -TL * t0 - TL * t1 + TL * t2 + TL * t3;
    const float T3 = -DARM * t0 + DARM * t1 - DARM * t2 + DARM * t3;

    const float sphi = __sinf(xr[0]), cphi = __cosf(xr[0]);
    const float sth  = __sinf(xr[1]), cth  = __cosf(xr[1]);
    const float spsi = __sinf(xr[2]), cpsi = __cosf(xr[2]);

    const float r0 = xr[6], r1 = xr[7], r2 = xr[8];
    const float v0 = xr[9], v1 = xr[10], v2 = xr[11];

    // ang_dot = inv(Mm) @ rate, analytic inverse (det(Mm) = cos(theta)):
    //   inv = [[1, s^2/cth, s*c/cth], [0, c, -s], [0, s/cth, c/cth]]  (s=sin phi, c=cos phi)
    const float rcth = 1.0f / cth;
    o[0] = r0 + (sphi * sphi * r1 + sphi * cphi * r2) * rcth;
    o[1] = cphi * r1 - sphi * r2;
    o[2] = (sphi * r1 + cphi * r2) * rcth;

    // vel passthrough
    o[3] = v0; o[4] = v1; o[5] = v2;

    // rate_dot = I_INV @ (torques[1:] - cross(rate, I@rate) - KR*rate)
    const float w0 = IXX * r0, w1 = IYY * r1, w2 = IZZ * r2;
    const float c0 = r1 * w2 - r2 * w1;
    const float c1 = r2 * w0 - r0 * w2;
    const float c2 = r0 * w1 - r1 * w0;
    o[6] = (T1 - c0 - KR * r0) * INV_IXX;
    o[7] = (T2 - c1 - KR * r1) * INV_IYY;
    o[8] = (T3 - c2 - KR * r2) * INV_IZZ;

    // vel_dot = rbi[:,2] * (T0/MASS) - KT*vel - GRAV
    const float ts = T0 / MASS;
    o[9]  = (cphi * cpsi * sth + sphi * spsi) * ts - KT * v0;
    o[10] = (cphi * spsi * sth - sphi * cpsi) * ts - KT * v1;
    o[11] = (cth * cphi) * ts - KT * v2 - GZ;

    o[12] = 0.0f; o[13] = 0.0f; o[14] = 0.0f; o[15] = 0.0f;
}

// ----- main kernel: async DMA in both directions, wave-private tiles -------
__global__ __launch_bounds__(THREADS) void quad_main_kernel(
    const float* __restrict__ x, const float* __restrict__ a,
    const float* __restrict__ b, const float* __restrict__ c,
    float* __restrict__ out) {
    __shared__ __align__(64) float tile[ROWS_PER_BLOCK * PAD_F];  // 10240 B

    const int tid  = threadIdx.x;
    const int lane = tid & 31;
    const int wave = tid >> 5;
    const long long blockRow0 = (long long)blockIdx.x * ROWS_PER_BLOCK;

    // Wave-private 2048 B region: rows [32*wave, 32*wave+32) of this block.
    const uint64_t gxbase =
        (uint64_t)(uintptr_t)x + (uint64_t)blockRow0 * 64ull + (uint64_t)wave * 2048ull;
    const uint64_t gobase =
        (uint64_t)(uintptr_t)out + (uint64_t)blockRow0 * 64ull + (uint64_t)wave * 2048ull;
    const uint32_t lds_wave_base = lds_off_u32(&tile[wave * 32 * PAD_F]);

    // Chunk <-> padded-row mapping, shared by load and store DMA.
    uint32_t goffs[4], loffs[4];
#pragma unroll
    for (int j = 0; j < 4; ++j) {
        const int chunk = lane + 32 * j;                        // 0..127 within wave tile
        goffs[j] = (uint32_t)chunk * 16u;                       // contiguous in global
        loffs[j] = lds_wave_base +
                   (uint32_t)(chunk >> 2) * (PAD_F * 4u) +      // row * 80B
                   (uint32_t)(chunk & 3) * 16u;                 // 16B col
    }

    // 4 contiguous B128 chunks per lane -> fully coalesced 2048 B per wave,
    // DMA'd into padded LDS rows (no VGPR round trip).
#pragma unroll
    for (int j = 0; j < 4; ++j) {
        asm volatile("global_load_async_to_lds_b128 %0, %1, %2"
                     :: "v"(loffs[j]), "v"(goffs[j]), "s"(gxbase)
                     : "memory");
    }
    // Only this wave's chunks feed this wave's rows: counter wait suffices.
    asm volatile("s_wait_asynccnt 0" ::: "memory");

    // Each lane reads its own padded row (2-way bank conflicts at most).
    float* rowp = &tile[tid * PAD_F];
    float xr[16];
#pragma unroll
    for (int k = 0; k < 4; ++k) {
        const f4 v = *(const f4*)(rowp + 4 * k);
        xr[4 * k + 0] = v.x; xr[4 * k + 1] = v.y;
        xr[4 * k + 2] = v.z; xr[4 * k + 3] = v.w;
    }

    float o[16];
    quad_dynamics(xr, a, b, c, o);

    // Write results back into the (now dead) input row of the LDS tile.
    // Lane L only touches row L; wave-private region -> no barrier.
#pragma unroll
    for (int k = 0; k < 4; ++k) {
        f4 v;
        v.x = o[4 * k + 0]; v.y = o[4 * k + 1];
        v.z = o[4 * k + 2]; v.w = o[4 * k + 3];
        *(f4*)(rowp + 4 * k) = v;
    }
    // LDS writes of all lanes in this wave must land before the DMA engine
    // reads them; DS ops cover all lanes, so a DS-counter wait is enough.
    asm volatile("s_wait_dscnt 0" ::: "memory");

    // Contiguous full-line async stores: each instruction drains 512 B of
    // consecutive global bytes from the padded LDS rows.  S_ENDPGM's implicit
    // wait-idle guarantees completion before LDS/wave teardown.
#pragma unroll
    for (int j = 0; j < 4; ++j) {
        asm volatile("global_store_async_from_lds_b128 %0, %1, %2"
                     :: "v"(goffs[j]), "v"(loffs[j]), "s"(gobase)
                     : "memory");
    }
}

// ----- tail kernel (generic remainder; grid 0 for the 2M case) -------------
__global__ void quad_tail_kernel(const float* __restrict__ x,
                                 const float* __restrict__ a,
                                 const float* __restrict__ b,
                                 const float* __restrict__ c,
                                 float* __restrict__ out,
                                 long long row0, long long nrows) {
    const long long i = row0 + (long long)blockIdx.x * blockDim.x + threadIdx.x;
    if (i >= nrows) return;
    const float* xrow = x + (size_t)i * ROW_F;
    float xr[16];
#pragma unroll
    for (int k = 0; k < 4; ++k) {
        const f4 v = *(const f4*)(xrow + 4 * k);
        xr[4 * k + 0] = v.x; xr[4 * k + 1] = v.y;
        xr[4 * k + 2] = v.z; xr[4 * k + 3] = v.w;
    }
    float o[16];
    quad_dynamics(xr, a, b, c, o);
    float* orow = out + (size_t)i * ROW_F;
#pragma unroll
    for (int k = 0; k < 4; ++k) {
        f4 v;
        v.x = o[4 * k + 0]; v.y = o[4 * k + 1];
        v.z = o[4 * k + 2]; v.w = o[4 * k + 3];
        *(f4*)(orow + 4 * k) = v;
    }
}

extern "C" void kernel_launch(void* const* d_in, const int* in_sizes, int n_in,
                              void* d_out, int out_size, void* d_ws, size_t ws_size,
                              hipStream_t stream) {
    // setup_inputs() order: t (int, unused), input, a, b, c
    const float* x = (const float*)d_in[1];
    const float* a = (const float*)d_in[2];
    const float* b = (const float*)d_in[3];
    const float* c = (const float*)d_in[4];
    float* out = (float*)d_out;

    const long long rows = (long long)in_sizes[1] / ROW_F;     // 2,000,000
    const long long fullBlocks = rows / ROWS_PER_BLOCK;        // 15,625 (exact)
    if (fullBlocks > 0) {
        quad_main_kernel<<<(unsigned)fullBlocks, THREADS, 0, stream>>>(x, a, b, c, out);
    }
    const long long rem = rows - fullBlocks * ROWS_PER_BLOCK;  // 0 for this shape
    if (rem > 0) {
        const unsigned tb = (unsigned)((rem + 127) / 128);
        quad_tail_kernel<<<tb, 128, 0, stream>>>(x, a, b, c, out,
                                                 fullBlocks * ROWS_PER_BLOCK, rows);
    }
}